// FFMLayer_16904991277364
// MI455X (gfx1250) — compile-verified
//
#include <hip/hip_runtime.h>

// ---------------------------------------------------------------------------
// FFM layer, reformulated:
//   A[i,j] = sum_f V[i, fd[j], f] * V[j, fd[i], f]   (fixed 2000x2000, symmetric)
//   out[s] = b0 + x_s.W + 0.5 * ( x_s^T A x_s - sum_n x_sn^2 A[n,n] )
// Kernel 1: build A pre-swizzled into WMMA B-fragment order (row pairs
//           interleaved: Asw[((k>>1)*NPAD + n)*2 + (k&1)]), plus diag(A),
//           padded W  -> d_ws
// Kernel 2: t = x@A via V_WMMA_F32_16X16X4_F32 (MT=64 samples/block, each
//           B-frag b64 load feeds 4 WMMAs), fused epilogue reduce
// Kernel 3: combine split partials + bias
// Workspace: (2000*2048 + 2048 + 2048 + 4096*4) floats ~= 15.7 MB
// ---------------------------------------------------------------------------

#define DIMD   2000
#define NPAD   2048
#define BATCH  4096
#define NFIELD 40
#define NF     8
#define MT     64            // samples per block (four 16-row WMMA sub-tiles)
#define NSUB   4             // MT / 16
#define CK     200           // K chunk staged in LDS (50 wmma k-steps)
#define NKC    10            // 2000 / 200
#define LDX    204           // padded LDS row stride (conflict-free, 16B aligned)
#define NSPLIT 4
#define NTPW   4             // N-tiles per wave: 128 / (NSPLIT*8)

typedef __attribute__((ext_vector_type(2))) float v2f;
typedef __attribute__((ext_vector_type(8))) float v8f;

// ---------------- Kernel 1: build swizzled A, diagA, padded W ---------------
__global__ __launch_bounds__(256) void FFM_build_A(
    const float* __restrict__ V, const int* __restrict__ fd,
    const float* __restrict__ W, float* __restrict__ Asw,
    float* __restrict__ dA, float* __restrict__ wp)
{
    const int idx = blockIdx.x * blockDim.x + threadIdx.x;   // < 2000*2048
    const int i = idx >> 11;          // K row 0..1999
    const int j = idx & (NPAD - 1);   // N col 0..2047
    float s = 0.0f;
    if (j < DIMD) {
        const int fi = fd[i];
        const int fj = fd[j];
        const float4* Vi = reinterpret_cast<const float4*>(V + (size_t)i * (NFIELD * NF) + fj * NF);
        const float4* Vj = reinterpret_cast<const float4*>(V + (size_t)j * (NFIELD * NF) + fi * NF);
        float4 a0 = Vi[0], a1 = Vi[1];
        float4 b0 = Vj[0], b1 = Vj[1];
        s = a0.x * b0.x + a0.y * b0.y + a0.z * b0.z + a0.w * b0.w
          + a1.x * b1.x + a1.y * b1.y + a1.z * b1.z + a1.w * b1.w;
    }
    // B-fragment-swizzled store: lane fragment {A[2p,n], A[2p+1,n]} contiguous
    Asw[(((size_t)(i >> 1) * NPAD + j) << 1) + (i & 1)] = s;
    if (i == j && j < DIMD) dA[j] = s;
    if (i == 0) {
        wp[j] = (j < DIMD) ? W[j] : 0.0f;
        if (j >= DIMD) dA[j] = 0.0f;
    }
}

// ---------------- Kernel 2: fused WMMA GEMM + reduction ---------------------
__global__ __launch_bounds__(256) void FFM_wmma_gemm(
    const float* __restrict__ x, const float* __restrict__ Asw,
    const float* __restrict__ dA, const float* __restrict__ wp,
    float* __restrict__ parts)
{
    extern __shared__ float lds[];                 // 64 * 204 floats = 52224 B
    const int tid  = threadIdx.x;
    const int w    = tid >> 5;                     // wave 0..7
    const int lane = tid & 31;
    const int hi   = lane >> 4;                    // 0 / 1 half-wave
    const int lo   = lane & 15;
    const int sp   = blockIdx.y;                   // N split 0..3
    const int s_base = blockIdx.x * MT;            // sample tile base

    v8f c[NTPW][NSUB];
    #pragma unroll
    for (int t = 0; t < NTPW; ++t)
        #pragma unroll
        for (int u = 0; u < NSUB; ++u)
            c[t][u] = (v8f){0.f, 0.f, 0.f, 0.f, 0.f, 0.f, 0.f, 0.f};

    for (int kc = 0; kc < NKC; ++kc) {
        // ---- stage x k-chunk into LDS (float4, padded stride) ----
        const int nv4 = MT * (CK / 4);             // 3200 float4
        for (int i = tid; i < nv4; i += 256) {
            const int r  = i / (CK / 4);
            const int cq = i % (CK / 4);
            float4 v = *reinterpret_cast<const float4*>(
                x + (size_t)(s_base + r) * DIMD + kc * CK + cq * 4);
            *reinterpret_cast<float4*>(&lds[r * LDX + cq * 4]) = v;
        }
        __syncthreads();

        for (int ks = 0; ks < CK / 4; ++ks) {
            const int k0 = ks * 4;                 // local k in chunk
            const int kg = kc * CK + k0;           // global k (multiple of 4)
            // A-operand (x tile, 16x4 f32): lanes 0-15 K={k0,k0+1}, 16-31 K={k0+2,k0+3}
            v2f a[NSUB];
            #pragma unroll
            for (int u = 0; u < NSUB; ++u) {
                const float* p = &lds[(u * 16 + lo) * LDX + k0 + hi * 2];
                a[u][0] = p[0]; a[u][1] = p[1];
            }
            #pragma unroll
            for (int t = 0; t < NTPW; ++t) {
                const int n = ((t * 32 + w * 4 + sp) * 16) + lo;
                // B-operand: contiguous pre-swizzled fragment -> one b64 load
                v2f bf = *reinterpret_cast<const v2f*>(
                    Asw + (((size_t)(kg >> 1) + hi) * NPAD + n) * 2);
                #pragma unroll
                for (int u = 0; u < NSUB; ++u)
                    c[t][u] = __builtin_amdgcn_wmma_f32_16x16x4_f32(
                        false, a[u], false, bf, (short)0, c[t][u], false, false);
            }
        }
        __syncthreads();
    }

    // ---- epilogue: per-lane accumulate x*(0.5*t + w_n) - 0.5*x^2*dA_n ----
    float racc[NSUB][8];
    #pragma unroll
    for (int u = 0; u < NSUB; ++u)
        #pragma unroll
        for (int v = 0; v < 8; ++v) racc[u][v] = 0.0f;

    #pragma unroll
    for (int t = 0; t < NTPW; ++t) {
        const int n = ((t * 32 + w * 4 + sp) * 16) + lo;
        const float wn = wp[n];
        const float dn = dA[n];
        const bool inb = (n < DIMD);
        #pragma unroll
        for (int u = 0; u < NSUB; ++u) {
            #pragma unroll
            for (int v = 0; v < 8; ++v) {
                const int r = v + hi * 8 + u * 16;       // sample row in tile
                float xv = inb ? x[(size_t)(s_base + r) * DIMD + n] : 0.0f;
                float tv = c[t][u][v];
                racc[u][v] += xv * (0.5f * tv + wn) - 0.5f * xv * xv * dn;
            }
        }
    }

    // ---- cross-lane reduce within 16-lane halves, stash per-wave in LDS ----
    #pragma unroll
    for (int u = 0; u < NSUB; ++u) {
        #pragma unroll
        for (int v = 0; v < 8; ++v) {
            float s = racc[u][v];
            s += __shfl_xor(s, 8, 32);
            s += __shfl_xor(s, 4, 32);
            s += __shfl_xor(s, 2, 32);
            s += __shfl_xor(s, 1, 32);
            if (lo == 0) {                          // lanes 0 and 16
                const int r = v + hi * 8 + u * 16;
                lds[w * MT + r] = s;                // LDS reused after final sync
            }
        }
    }
    __syncthreads();

    if (tid < MT) {
        float p = 0.0f;
        #pragma unroll
        for (int w2 = 0; w2 < 8; ++w2) p += lds[w2 * MT + tid];
        parts[(size_t)(s_base + tid) * NSPLIT + sp] = p;
    }
}

// ---------------- Kernel 3: combine splits + bias ---------------------------
__global__ __launch_bounds__(256) void FFM_finalize(
    const float* __restrict__ parts, const float* __restrict__ b,
    float* __restrict__ out)
{
    const int i = blockIdx.x * blockDim.x + threadIdx.x;
    if (i < BATCH) {
        float p = b[0];
        #pragma unroll
        for (int s = 0; s < NSPLIT; ++s) p += parts[(size_t)i * NSPLIT + s];
        out[i] = p;
    }
}

// ---------------------------------------------------------------------------
extern "C" void kernel_launch(void* const* d_in, const int* in_sizes, int n_in,
                              void* d_out, int out_size, void* d_ws, size_t ws_size,
                              hipStream_t stream)
{
    const float* x  = (const float*)d_in[0];   // (B, D)
    const int*   fd = (const int*)  d_in[1];   // (D,)
    const float* b  = (const float*)d_in[2];   // (1,)
    const float* W  = (const float*)d_in[3];   // (D, 1)
    const float* V  = (const float*)d_in[4];   // (D, FIELDS, F)
    float* out = (float*)d_out;

    float* Asw   = (float*)d_ws;               // 2000*2048 floats (swizzled)
    float* dA    = Asw + (size_t)DIMD * NPAD;  // 2048
    float* wp    = dA + NPAD;                  // 2048
    float* parts = wp + NPAD;                  // 4096*4

    // Kernel 1: 2000*2048 elements / 256 = 16000 blocks
    FFM_build_A<<<16000, 256, 0, stream>>>(V, fd, W, Asw, dA, wp);

    // Kernel 2: (B/64) x NSPLIT blocks, 8 waves each, 52224 B dynamic LDS
    dim3 g2(BATCH / MT, NSPLIT);
    FFM_wmma_gemm<<<g2, 256, MT * LDX * sizeof(float), stream>>>(x, Asw, dA, wp, parts);

    // Kernel 3
    FFM_finalize<<<BATCH / 256, 256, 0, stream>>>(parts, b, out);
}